// GNNLayerF_37409165148998
// MI455X (gfx1250) — compile-verified
//
#include <hip/hip_runtime.h>

// GNN layer: out = relu(x@W1^T + scatter_add(x@W2^T gathered by col, into row))
// x: [100000,128] f32, edge_index: [2,1600000] int32 (JAX w/o x64 -> int32),
// W1,W2: [128,128] f32.  Output: [100000,128] f32.

typedef __attribute__((ext_vector_type(2))) float v2f;
typedef __attribute__((ext_vector_type(8))) float v8f;

#define DFEAT   128
#define KSTEPS  (DFEAT / 4)   // 32 k-steps of V_WMMA_F32_16X16X4_F32
#define WSTRIDE 132           // padded LDS row stride (floats) -> conflict-free
                              // fragment reads: banks 4*mr+4s+{0..3} cover all 64
#define WMAT_FLOATS (DFEAT * WSTRIDE)        // 16896 floats per weight matrix
#define ATILE_FLOATS (16 * WSTRIDE)          // 2112 floats per A tile
#define LDS_FLOATS (2 * WMAT_FLOATS + ATILE_FLOATS)
#define LDS_BYTES  (LDS_FLOATS * 4)          // 143,616 B -> 2 blocks / WGP

// ---------------------------------------------------------------------------
// Kernel 1: dual GEMM via V_WMMA_F32_16X16X4_F32 with LDS-resident weights
// and a block-shared LDS A tile.
//   - W1+W2 staged once per block (padded layout, conflict-free ds_load_b64)
//   - A tile (16x128) staged per row tile, shared by all 8 waves
//   - inner loop: 3x ds_load_b64 + 2x wmma per k-step, no global/scratch
//   selfOut  = x @ W1^T  -> d_out (accumulated into by scatter)
//   neighOut = x @ W2^T  -> workspace (gathered by edges)
// EXEC is all-ones wherever WMMA executes (tile guard is scalar).
// ---------------------------------------------------------------------------
__global__ __launch_bounds__(256) void gemm2_wmma_f32(
    const float* __restrict__ x,
    const float* __restrict__ W1,
    const float* __restrict__ W2,
    float* __restrict__ selfOut,
    float* __restrict__ neighOut,
    int nTiles)
{
    extern __shared__ float lds[];
    float* sW1 = lds;
    float* sW2 = lds + WMAT_FLOATS;
    float* sA  = lds + 2 * WMAT_FLOATS;

    const int lane = threadIdx.x;        // 0..31
    const int wv   = threadIdx.y;        // 0..7
    const int tid  = wv * 32 + lane;     // 0..255
    const int n0   = wv * 16;            // col tile base (wave id * 16)
    const int mr   = lane & 15;
    const int half = lane >> 4;          // 0: K={0,1}, 1: K={2,3} within a step

    // ---- stage W1/W2 into padded LDS (float4 copies, once per block) ----
#pragma unroll
    for (int i = 0; i < 16; ++i) {
        const int i4  = tid + i * 256;       // 0..4095 float4 index
        const int row = i4 >> 5;             // 32 float4 per row
        const int c   = (i4 & 31) << 2;      // column in floats
        *(float4*)(sW1 + row * WSTRIDE + c) = ((const float4*)W1)[i4];
        *(float4*)(sW2 + row * WSTRIDE + c) = ((const float4*)W2)[i4];
    }
    __syncthreads();

    // Fragment read offsets (8B ds loads, conflict-free with WSTRIDE=132):
    //   A : lane -> sA [ mr*132        + 4s + 2*half ]
    //   B1: lane -> sW1[(n0+mr)*132    + 4s + 2*half ]   (B[k][n] = W[n][k])
    const float* aFrag  = sA  + mr * WSTRIDE        + 2 * half;
    const float* b1Frag = sW1 + (n0 + mr) * WSTRIDE + 2 * half;
    const float* b2Frag = sW2 + (n0 + mr) * WSTRIDE + 2 * half;

    // ---- grid-stride over 16-row tiles ----
    for (int t = blockIdx.x; t < nTiles; t += gridDim.x) {
        const int m0 = t * 16;

        // stage A tile (16 rows x 128 floats = 512 float4; 2 per thread)
        const float4* xt = (const float4*)(x + (size_t)m0 * DFEAT);
#pragma unroll
        for (int i = 0; i < 2; ++i) {
            const int i4  = tid + i * 256;   // 0..511
            const int row = i4 >> 5;
            const int c   = (i4 & 31) << 2;
            *(float4*)(sA + row * WSTRIDE + c) = xt[i4];
        }
        __syncthreads();

        v8f acc1 = {};
        v8f acc2 = {};
#pragma unroll 8
        for (int s = 0; s < KSTEPS; ++s) {
            v2f a  = *(const v2f*)(aFrag  + 4 * s);
            v2f b1 = *(const v2f*)(b1Frag + 4 * s);
            v2f b2 = *(const v2f*)(b2Frag + 4 * s);
            // (neg_a, A, neg_b, B, c_mod, C, reuse_a, reuse_b)
            acc1 = __builtin_amdgcn_wmma_f32_16x16x4_f32(
                       false, a, false, b1, (short)0, acc1, false, false);
            acc2 = __builtin_amdgcn_wmma_f32_16x16x4_f32(
                       false, a, false, b2, (short)0, acc2, false, false);
        }

        // D layout: VGPR v -> row m0 + v + 8*half, col n0 + mr
#pragma unroll
        for (int v = 0; v < 8; ++v) {
            const size_t m = (size_t)(m0 + v + 8 * half);
            const int    n = n0 + mr;
            selfOut [m * DFEAT + n] = acc1[v];
            neighOut[m * DFEAT + n] = acc2[v];
        }
        __syncthreads();   // protect sA before next iteration's staging
    }
}

// ---------------------------------------------------------------------------
// Kernel 2: edge scatter-add. One wave per edge; lane l moves float4
// T[col][4l..4l+3] (coalesced 512B per wave) and atomically adds into
// out[row][4l..4l+3].  Both 51.2 MB tables are L2-resident (192 MB L2), so
// the random gathers and the 205M global_atomic_add_f32 run at L2 speed.
// ---------------------------------------------------------------------------
__global__ __launch_bounds__(256) void scatter_add_edges(
    const int* __restrict__ rows,   // destinations (edge_index[0])
    const int* __restrict__ cols,   // sources      (edge_index[1])
    const float* __restrict__ T,    // neighbor features [N,128]
    float* __restrict__ out,        // accumulator [N,128] (holds x@W1^T)
    int nEdges)
{
    const int gwave = (int)((blockIdx.x * 256u + threadIdx.x) >> 5);
    const int lane  = threadIdx.x & 31;
    if (gwave >= nEdges) return;

    const int r = rows[gwave];
    const int c = cols[gwave];

    const float4 v = *((const float4*)(T + (size_t)c * DFEAT) + lane);
    float* dst = out + (size_t)r * DFEAT + 4 * lane;

    unsafeAtomicAdd(dst + 0, v.x);   // -> global_atomic_add_f32
    unsafeAtomicAdd(dst + 1, v.y);
    unsafeAtomicAdd(dst + 2, v.z);
    unsafeAtomicAdd(dst + 3, v.w);
}

// ---------------------------------------------------------------------------
// Kernel 3: in-place ReLU over the accumulated output (float4 vectorized).
// ---------------------------------------------------------------------------
__global__ __launch_bounds__(256) void relu_inplace(float4* __restrict__ out, int n4)
{
    const int i = blockIdx.x * 256 + threadIdx.x;
    if (i < n4) {
        float4 v = out[i];
        v.x = fmaxf(v.x, 0.0f);
        v.y = fmaxf(v.y, 0.0f);
        v.z = fmaxf(v.z, 0.0f);
        v.w = fmaxf(v.w, 0.0f);
        out[i] = v;
    }
}

extern "C" void kernel_launch(void* const* d_in, const int* in_sizes, int n_in,
                              void* d_out, int out_size, void* d_ws, size_t ws_size,
                              hipStream_t stream)
{
    const float* x   = (const float*)d_in[0];
    const int*   ei  = (const int*)  d_in[1];   // [2, nEdges] flat, int32
    const float* W1  = (const float*)d_in[2];
    const float* W2  = (const float*)d_in[3];
    float*       out = (float*)d_out;
    float*       T   = (float*)d_ws;            // [nNodes, 128] scratch

    const int nNodes = in_sizes[0] / DFEAT;     // 100000 (divisible by 16)
    const int nEdges = in_sizes[1] / 2;         // 1600000
    const int nTiles = nNodes / 16;             // 6250

    // Allow >64KB dynamic LDS (AMD hw max is 320KB/WGP; we use ~141KB/block).
    (void)hipFuncSetAttribute((const void*)gemm2_wmma_f32,
                              hipFuncAttributeMaxDynamicSharedMemorySize,
                              (int)LDS_BYTES);

    // 1) selfOut -> d_out, neighOut -> workspace (both fully overwritten).
    //    625 blocks x 8 waves; each block grid-strides over ~10 row tiles so
    //    the one-time LDS weight staging is amortized.
    gemm2_wmma_f32<<<dim3(625), dim3(32, 8), LDS_BYTES, stream>>>(x, W1, W2, out, T, nTiles);

    // 2) atomic scatter-add of gathered neighbor features into d_out
    const int scatterBlocks = (nEdges + 7) / 8;           // 8 edges (waves) / block
    scatter_add_edges<<<scatterBlocks, 256, 0, stream>>>(ei, ei + nEdges, T, out, nEdges);

    // 3) relu in place
    const int n4 = nNodes * DFEAT / 4;
    relu_inplace<<<(n4 + 255) / 256, 256, 0, stream>>>((float4*)out, n4);
}